// PostAttentionPruner_75617194213659
// MI455X (gfx1250) — compile-verified
//
#include <hip/hip_runtime.h>
#include <hip/hip_bf16.h>

typedef __attribute__((ext_vector_type(2))) float v2f;
typedef __attribute__((ext_vector_type(8))) float v8f;

#define EPSF 1e-10f

// ---------------------------------------------------------------------------
// Pass 1 over edges: edge_attn_sum = mean over 8 heads; deterministic
// fixed-point scatter-add to destination nodes; block-reduced max of edge sums.
// E is a multiple of 256 (1,600,000), no tail guard needed.
// ---------------------------------------------------------------------------
__global__ __launch_bounds__(256) void edge_pass1_kernel(
    const float* __restrict__ naw, const int* __restrict__ eidx_dst,
    float* __restrict__ edge_sum, unsigned long long* __restrict__ agg_q,
    unsigned* __restrict__ emax_bits)
{
    int e = blockIdx.x * 256 + threadIdx.x;
    const float4* p = (const float4*)(naw + (size_t)e * 8);
    float4 a = p[0], b = p[1];
    float s = (a.x + a.y + a.z + a.w + b.x + b.y + b.z + b.w) * 0.125f;
    edge_sum[e] = s;
    // fixed point (value in [0,1)): exact, order-independent accumulation
    unsigned long long q = (unsigned long long)((double)s * 4294967296.0);
    atomicAdd(&agg_q[eidx_dst[e]], q);

    __shared__ float sm[256];
    sm[threadIdx.x] = s;
    __syncthreads();
    for (int off = 128; off > 0; off >>= 1) {
        if (threadIdx.x < off) sm[threadIdx.x] = fmaxf(sm[threadIdx.x], sm[threadIdx.x + off]);
        __syncthreads();
    }
    if (threadIdx.x == 0) atomicMax(emax_bits, __float_as_uint(sm[0]));
}

// Convert fixed-point node aggregate back to float; reduce max over nodes.
__global__ __launch_bounds__(256) void node_max_kernel(
    const unsigned long long* __restrict__ agg_q, float* __restrict__ agg_f,
    unsigned* __restrict__ nmax_bits, int n)
{
    int i = blockIdx.x * 256 + threadIdx.x;
    float v = 0.0f;
    if (i < n) {
        v = (float)((double)agg_q[i] * 2.3283064365386963e-10); // * 2^-32
        agg_f[i] = v;
    }
    __shared__ float sm[256];
    sm[threadIdx.x] = v;
    __syncthreads();
    for (int off = 128; off > 0; off >>= 1) {
        if (threadIdx.x < off) sm[threadIdx.x] = fmaxf(sm[threadIdx.x], sm[threadIdx.x + off]);
        __syncthreads();
    }
    if (threadIdx.x == 0) atomicMax(nmax_bits, __float_as_uint(sm[0]));
}

// ---------------------------------------------------------------------------
// Node gate: sigmoid( nf[i,0:256] . w[0:256] + w[256]*agg_norm[i] + b )
// 32 rows / block (2 waves x 16 rows), K=256 in 64 WMMA f32 16x16x4 chunks.
// LDS pitch 260 floats -> conflict-free ds_load_b64 fragment reads.
// N is a multiple of 32 (100,000 / 32 = 3125), no tail guard.
// ---------------------------------------------------------------------------
__global__ __launch_bounds__(64) void node_scores_kernel(
    const float* __restrict__ nf, const float* __restrict__ w,
    const float* __restrict__ bias, const float* __restrict__ agg_f,
    const unsigned* __restrict__ nmax_bits, float* __restrict__ out)
{
    __shared__ float sA[32 * 260];
    __shared__ float sw[257];
    int t = threadIdx.x;
    int row0 = blockIdx.x * 32;

    for (int i = t; i < 257; i += 64) sw[i] = w[i];
    // stage 32 rows x 256 floats = 2048 float4, fully coalesced b128 loads
    for (int q = t; q < 2048; q += 64) {
        int r = q >> 6, c4 = q & 63;
        float4 v = ((const float4*)(nf + (size_t)(row0 + r) * 256))[c4];
        *(float4*)&sA[r * 260 + c4 * 4] = v;
    }
    __syncthreads();

    int wave = t >> 5, lane = t & 31;
    int half = lane >> 4, m = lane & 15;
    const float* rowp = &sA[(wave * 16 + m) * 260];

    v8f c = {0.f, 0.f, 0.f, 0.f, 0.f, 0.f, 0.f, 0.f};
#pragma unroll
    for (int k = 0; k < 256; k += 4) {
        int off = k + 2 * half;
        v2f av = {sw[off], sw[off + 1]};      // A: weights replicated across M
        v2f bv = {rowp[off], rowp[off + 1]};  // B: feature rows as columns
        c = __builtin_amdgcn_wmma_f32_16x16x4_f32(false, av, false, bv,
                                                  (short)0, c, false, false);
    }
    if (half == 0) {
        int i = row0 + wave * 16 + m;  // dot for this row is in c[0], lanes 0..15
        float nm = __uint_as_float(*nmax_bits);
        float imp = agg_f[i] / (nm + EPSF);
        float x = c[0] + sw[256] * imp + bias[0];
        out[i] = 1.0f / (1.0f + __expf(-x));
    }
}

// ---------------------------------------------------------------------------
// Edge gate: sigmoid( ef[e,0:128] . w[0:128] + w[128]*imp_norm[e] + b )
// 64 rows / block (4 waves x 16 rows), K=128 in 32 WMMA chunks, pitch 132.
// E is a multiple of 64 (1,600,000 / 64 = 25,000), no tail guard.
// ---------------------------------------------------------------------------
__global__ __launch_bounds__(128) void edge_scores_kernel(
    const float* __restrict__ ef, const float* __restrict__ w,
    const float* __restrict__ bias, const float* __restrict__ edge_sum,
    const unsigned* __restrict__ emax_bits, float* __restrict__ out)
{
    __shared__ float sA[64 * 132];
    __shared__ float sw[129];
    int t = threadIdx.x;
    int row0 = blockIdx.x * 64;

    for (int i = t; i < 129; i += 128) sw[i] = w[i];
    // stage 64 rows x 128 floats = 2048 float4, coalesced
    for (int q = t; q < 2048; q += 128) {
        int r = q >> 5, c4 = q & 31;
        float4 v = ((const float4*)(ef + (size_t)(row0 + r) * 128))[c4];
        *(float4*)&sA[r * 132 + c4 * 4] = v;
    }
    __syncthreads();

    int wave = t >> 5, lane = t & 31;
    int half = lane >> 4, m = lane & 15;
    const float* rowp = &sA[(wave * 16 + m) * 132];

    v8f c = {0.f, 0.f, 0.f, 0.f, 0.f, 0.f, 0.f, 0.f};
#pragma unroll
    for (int k = 0; k < 128; k += 4) {
        int off = k + 2 * half;
        v2f av = {sw[off], sw[off + 1]};
        v2f bv = {rowp[off], rowp[off + 1]};
        c = __builtin_amdgcn_wmma_f32_16x16x4_f32(false, av, false, bv,
                                                  (short)0, c, false, false);
    }
    if (half == 0) {
        int e = row0 + wave * 16 + m;
        float em = __uint_as_float(*emax_bits);
        float imp = edge_sum[e] / (em + EPSF);
        float x = c[0] + sw[128] * imp + bias[0];
        out[e] = 1.0f / (1.0f + __expf(-x));
    }
}

// ---------------------------------------------------------------------------
// Exact top-K via 3-pass radix select on float bit patterns (all scores > 0).
// state: [0]=prefix  [1]=remaining  [2]=threshold key  [3]=need (ties to take)
// ---------------------------------------------------------------------------
__global__ __launch_bounds__(256) void radix_hist_kernel(
    const float* __restrict__ scores, int n, unsigned* __restrict__ hist,
    const unsigned* __restrict__ state, int pass)
{
    __shared__ unsigned lh[2048];
    for (int i = threadIdx.x; i < 2048; i += blockDim.x) lh[i] = 0;
    __syncthreads();
    unsigned prefix = state[0];
    for (int i = blockIdx.x * blockDim.x + threadIdx.x; i < n;
         i += gridDim.x * blockDim.x) {
        unsigned key = __float_as_uint(scores[i]);
        if (pass == 0) {
            atomicAdd(&lh[key >> 21], 1u);
        } else if (pass == 1) {
            if ((key >> 21) == prefix) atomicAdd(&lh[(key >> 10) & 2047u], 1u);
        } else {
            if ((key >> 10) == prefix) atomicAdd(&lh[key & 1023u], 1u);
        }
    }
    __syncthreads();
    for (int i = threadIdx.x; i < 2048; i += blockDim.x)
        if (lh[i]) atomicAdd(&hist[i], lh[i]);
}

__global__ void radix_pick_kernel(unsigned* hist, unsigned* state, int pass,
                                  unsigned K)
{
    unsigned remaining = (pass == 0) ? K : state[1];
    int nb = (pass == 2) ? 1024 : 2048;
    unsigned acc = 0, rem_after = remaining;
    int sel = 0;
    bool found = false;
    for (int b = nb - 1; b >= 0; --b) {
        unsigned cnt = hist[b];
        hist[b] = 0;  // re-zero for next pass / next selection / graph replay
        if (!found) {
            if (acc + cnt >= remaining) { sel = b; found = true; rem_after = remaining - acc; }
            else acc += cnt;
        }
    }
    if (pass == 0)      { state[0] = (unsigned)sel;                    state[1] = rem_after; }
    else if (pass == 1) { state[0] = (state[0] << 11) | (unsigned)sel; state[1] = rem_after; }
    else                { state[2] = (state[0] << 10) | (unsigned)sel; state[3] = rem_after; }
}

__global__ __launch_bounds__(1024) void eq_blockcount_kernel(
    const float* __restrict__ scores, int n, const unsigned* __restrict__ state,
    unsigned* __restrict__ bsums)
{
    int i = blockIdx.x * 1024 + threadIdx.x;
    unsigned T = state[2];
    unsigned v = (i < n && __float_as_uint(scores[i]) == T) ? 1u : 0u;
    __shared__ unsigned s[1024];
    s[threadIdx.x] = v;
    __syncthreads();
    for (int off = 512; off > 0; off >>= 1) {
        if (threadIdx.x < off) s[threadIdx.x] += s[threadIdx.x + off];
        __syncthreads();
    }
    if (threadIdx.x == 0) bsums[blockIdx.x] = s[0];
}

__global__ void scan_bsums_kernel(unsigned* bsums, int nblocks)
{
    unsigned acc = 0;
    for (int i = 0; i < nblocks; ++i) {
        unsigned t = bsums[i];
        bsums[i] = acc;
        acc += t;
    }
}

__global__ __launch_bounds__(1024) void write_mask_kernel(
    const float* __restrict__ scores, int n, const unsigned* __restrict__ state,
    const unsigned* __restrict__ bsums, float* __restrict__ mask)
{
    __shared__ unsigned s[1024];
    int t = threadIdx.x;
    int i = blockIdx.x * 1024 + t;
    unsigned T = state[2], need = state[3];
    unsigned key = (i < n) ? __float_as_uint(scores[i]) : 0u;
    unsigned eq = (i < n && key == T) ? 1u : 0u;
    s[t] = eq;
    __syncthreads();
    for (int off = 1; off < 1024; off <<= 1) {  // Hillis-Steele inclusive scan
        unsigned u = (t >= off) ? s[t - off] : 0u;
        __syncthreads();
        s[t] += u;
        __syncthreads();
    }
    unsigned excl = s[t] - eq;
    if (i < n) {
        float m = 0.0f;
        if (key > T) m = 1.0f;
        else if (eq && (bsums[blockIdx.x] + excl) < need) m = 1.0f;  // index-order ties
        mask[i] = m;
    }
}

// ---------------------------------------------------------------------------
extern "C" void kernel_launch(void* const* d_in, const int* in_sizes, int n_in,
                              void* d_out, int out_size, void* d_ws, size_t ws_size,
                              hipStream_t stream)
{
    const float* nf  = (const float*)d_in[0];
    const float* ef  = (const float*)d_in[1];
    const float* naw = (const float*)d_in[2];
    // d_in[3] (edge_attn_weights) is unused by the reference
    const float* ngw = (const float*)d_in[4];
    const float* ngb = (const float*)d_in[5];
    const float* egw = (const float*)d_in[6];
    const float* egb = (const float*)d_in[7];
    const int*  eidx = (const int*)d_in[8];

    const int N = in_sizes[0] / 256;   // 100,000
    const int E = in_sizes[1] / 128;   // 1,600,000
    const unsigned NODE_K = (unsigned)((N * 7) / 10);  // 70,000
    const unsigned EDGE_K = (unsigned)((E * 7) / 10);  // 1,120,000

    float* out = (float*)d_out;
    float* node_scores = out;
    float* edge_scores = out + N;
    float* node_mask   = out + N + E;
    float* edge_mask   = out + N + E + N;

    // workspace layout
    char* base = (char*)d_ws;
    float* edge_sum = (float*)base;                               // E floats
    size_t off = (size_t)E * 4;
    unsigned long long* agg_q = (unsigned long long*)(base + off); // N u64
    off += (size_t)N * 8;
    float* agg_f = (float*)(base + off);                           // N floats
    off += (size_t)N * 4;
    unsigned* hist = (unsigned*)(base + off);                      // 2048 u32
    size_t zero_off = off;
    off += 2048 * 4;
    unsigned* state = (unsigned*)(base + off);                     // 8 u32
    off += 8 * 4;
    unsigned* maxbits = (unsigned*)(base + off);                   // [0]=edge [1]=node
    off += 2 * 4;
    size_t zero_len = off - zero_off;
    unsigned* bsums = (unsigned*)(base + off);                     // 1600 u32

    hipMemsetAsync(agg_q, 0, (size_t)N * 8, stream);
    hipMemsetAsync(base + zero_off, 0, zero_len, stream);

    // --- importance ---
    edge_pass1_kernel<<<E / 256, 256, 0, stream>>>(naw, eidx + E, edge_sum, agg_q,
                                                   &maxbits[0]);
    node_max_kernel<<<(N + 255) / 256, 256, 0, stream>>>(agg_q, agg_f, &maxbits[1], N);

    // --- gated scores (WMMA f32 16x16x4) ---
    node_scores_kernel<<<N / 32, 64, 0, stream>>>(nf, ngw, ngb, agg_f, &maxbits[1],
                                                  node_scores);
    edge_scores_kernel<<<E / 64, 128, 0, stream>>>(ef, egw, egb, edge_sum, &maxbits[0],
                                                   edge_scores);

    // --- exact top-K masks via radix select ---
    // nodes
    {
        int hgrid = (N + 255) / 256; if (hgrid > 2048) hgrid = 2048;
        for (int p = 0; p < 3; ++p) {
            radix_hist_kernel<<<hgrid, 256, 0, stream>>>(node_scores, N, hist, state, p);
            radix_pick_kernel<<<1, 1, 0, stream>>>(hist, state, p, NODE_K);
        }
        int nb = (N + 1023) / 1024;
        eq_blockcount_kernel<<<nb, 1024, 0, stream>>>(node_scores, N, state, bsums);
        scan_bsums_kernel<<<1, 1, 0, stream>>>(bsums, nb);
        write_mask_kernel<<<nb, 1024, 0, stream>>>(node_scores, N, state, bsums, node_mask);
    }
    // edges
    {
        int hgrid = (E + 255) / 256; if (hgrid > 2048) hgrid = 2048;
        for (int p = 0; p < 3; ++p) {
            radix_hist_kernel<<<hgrid, 256, 0, stream>>>(edge_scores, E, hist, state, p);
            radix_pick_kernel<<<1, 1, 0, stream>>>(hist, state, p, EDGE_K);
        }
        int nb = (E + 1023) / 1024;
        eq_blockcount_kernel<<<nb, 1024, 0, stream>>>(edge_scores, E, state, bsums);
        scan_bsums_kernel<<<1, 1, 0, stream>>>(bsums, nb);
        write_mask_kernel<<<nb, 1024, 0, stream>>>(edge_scores, E, state, bsums, edge_mask);
    }
}